// Attention_16183436771581
// MI455X (gfx1250) — compile-verified
//
#include <hip/hip_runtime.h>
#include <hip/hip_bf16.h>

typedef float v2f __attribute__((ext_vector_type(2)));
typedef float v8f __attribute__((ext_vector_type(8)));

#define DIM        4096
#define N_HEADS    32
#define N_KV_HEADS 8
#define HEAD_DIM   128
#define N_REP      4
#define MAX_BS     16
#define KV_LEN     4096   // start_pos (4095) + seqlen (1)
#define TILE       512
#define BLK        256
#define GSPLIT     4      // K-split inside each GEMM block
#define ASPLIT     4      // KV-split across attention blocks
#define CHUNK      (KV_LEN / ASPLIT)   // 1024

// ---------------------------------------------------------------------------
// GEMM: C(16 x N) = A(16 x K) * B(K x N), all fp32 row-major. K is a
// template constant so the split-K inner loop has a uniform, compile-time
// trip count (K/GSPLIT/4) -> compiler unrolls and pipelines loads ahead of
// V_WMMA_F32_16X16X4_F32. One block (4 waves) = one 16-col tile, each wave
// covers K/4; partials combined via LDS in fixed order. Grid = N/16 blocks.
// ---------------------------------------------------------------------------
template <int K>
__global__ __launch_bounds__(128)
void gemm16_wmma(const float* __restrict__ A,
                 const float* __restrict__ B,
                 float* __restrict__ C,
                 int N)
{
    constexpr int KCHUNK = K / GSPLIT;   // 1024

    __shared__ float part[GSPLIT * 256];

    const int wave = threadIdx.x >> 5;   // k-split id, 0..3
    const int lane = threadIdx.x & 31;
    const int n0   = blockIdx.x * 16;

    const int m     = lane & 15;         // M for A, N for B
    const int khalf = (lane >> 4) * 2;   // lanes 0-15 -> K={0,1}; 16-31 -> K={2,3}
    const int kbeg  = wave * KCHUNK;

    // pre-offset bases by this wave's K chunk; loop bounds stay uniform consts
    const float* __restrict__ arow = A + (size_t)m * K + khalf + kbeg;
    const float* __restrict__ bbas = B + ((size_t)khalf + kbeg) * N + n0 + m;

    v8f acc = {};

#pragma unroll 8
    for (int kk = 0; kk < KCHUNK; kk += 4) {
        float2 af = *(const float2*)(arow + kk);       // A[m][kbeg+kk+khalf+{0,1}]
        v2f a; a[0] = af.x; a[1] = af.y;
        v2f b;
        b[0] = bbas[(size_t)kk * N];                   // B[kbeg+kk+khalf  ][n0+m]
        b[1] = bbas[(size_t)(kk + 1) * N];             // B[kbeg+kk+khalf+1][n0+m]
        acc = __builtin_amdgcn_wmma_f32_16x16x4_f32(
            false, a, false, b, (short)0, acc, false, false);
    }

    // stash this wave's partial tile: element index e = r*32 + lane
#pragma unroll
    for (int r = 0; r < 8; ++r)
        part[wave * 256 + r * 32 + lane] = acc[r];
    __syncthreads();

    // deterministic combine (w = 0..3), 2 elements per thread
    for (int e = threadIdx.x; e < 256; e += 128) {
        float s = part[e] + part[256 + e] + part[512 + e] + part[768 + e];
        int r  = e >> 5;
        int ln = e & 31;
        int mm = r + ((ln >> 4) * 8);      // D layout: lanes>=16 hold M=8..15
        int nn = n0 + (ln & 15);
        C[(size_t)mm * N + nn] = s;
    }
}

// ---------------------------------------------------------------------------
// RoPE on q: xq is 16 x (32*128). One thread per (b, h, i): 32768 threads.
// ---------------------------------------------------------------------------
__global__ __launch_bounds__(256)
void rope_q_kernel(float* __restrict__ xq,
                   const float* __restrict__ fcos,
                   const float* __restrict__ fsin)
{
    int tid = blockIdx.x * blockDim.x + threadIdx.x;   // < 32768
    int i   = tid & 63;
    int bh  = tid >> 6;
    float c = fcos[i], s = fsin[i];
    float* p = xq + (size_t)bh * HEAD_DIM + 2 * i;
    float e = p[0], o = p[1];
    p[0] = e * c - o * s;
    p[1] = e * s + o * c;
}

// RoPE on new k: xk is 16 x (8*128). 8192 threads.
__global__ __launch_bounds__(256)
void rope_k_kernel(float* __restrict__ xk,
                   const float* __restrict__ fcos,
                   const float* __restrict__ fsin)
{
    int tid = blockIdx.x * blockDim.x + threadIdx.x;   // < 8192
    int i   = tid & 63;
    int bg  = tid >> 6;
    float c = fcos[i], s = fsin[i];
    float* p = xk + (size_t)bg * HEAD_DIM + 2 * i;
    float e = p[0], o = p[1];
    p[0] = e * c - o * s;
    p[1] = e * s + o * c;
}

// ---------------------------------------------------------------------------
// Flash-decode partial: block = ((b*8+g)*ASPLIT + sp). Chunk-local online
// softmax over CHUNK positions; emits unnormalized (acc, m, l) partials.
// Position 4095 (new token) sourced from rope'd xk/xv workspace.
// ---------------------------------------------------------------------------
__global__ __launch_bounds__(BLK)
void attn_partial(const float* __restrict__ xq,       // 16 x 4096 (rope'd)
                  const float* __restrict__ xk,       // 16 x 1024 (rope'd new k)
                  const float* __restrict__ xv,       // 16 x 1024 (new v)
                  const float* __restrict__ cache_k,  // 16 x 4096 x 8 x 128
                  const float* __restrict__ cache_v,
                  float* __restrict__ pacc,           // [bg][sp][h][d]
                  float* __restrict__ pm,             // [bg][sp][h]
                  float* __restrict__ pl)             // [bg][sp][h]
{
    const int sp = blockIdx.x & (ASPLIT - 1);
    const int bg = blockIdx.x >> 2;            // b*8 + g
    const int b  = bg >> 3;
    const int g  = bg & 7;
    const int t  = threadIdx.x;
    const int chunk0 = sp * CHUNK;

    __shared__ float q_s[N_REP * HEAD_DIM];
    __shared__ float s_s[N_REP * TILE];
    __shared__ float red[N_REP * BLK];
    __shared__ float tile_m[N_REP], corr[N_REP], run_m[N_REP], run_l[N_REP];

    for (int idx = t; idx < N_REP * HEAD_DIM; idx += BLK)
        q_s[idx] = xq[(size_t)b * DIM + g * (N_REP * HEAD_DIM) + idx];
    if (t < N_REP) { run_m[t] = -1e30f; run_l[t] = 0.0f; }
    __syncthreads();

    const int d  = t & 127;
    const int hp = t >> 7;
    float acc0 = 0.0f, acc1 = 0.0f;

    const float scale = 0.08838834764831845f;  // 1/sqrt(128)
    const float* knew = xk + ((size_t)b * N_KV_HEADS + g) * HEAD_DIM;
    const float* vnew = xv + ((size_t)b * N_KV_HEADS + g) * HEAD_DIM;

    for (int tile = chunk0; tile < chunk0 + CHUNK; tile += TILE) {
        // ---- raw scores ----
        float lm0 = -1e30f, lm1 = -1e30f, lm2 = -1e30f, lm3 = -1e30f;
        for (int pp = t; pp < TILE; pp += BLK) {
            int pos = tile + pp;
            const float* kptr = (pos == KV_LEN - 1)
                ? knew
                : (cache_k + (((size_t)b * KV_LEN + pos) * N_KV_HEADS + g) * HEAD_DIM);
            if (pp + BLK < TILE) {   // prefetch next K row for this thread
                int np = pos + BLK;
                const float* nk = (np == KV_LEN - 1)
                    ? knew
                    : (cache_k + (((size_t)b * KV_LEN + np) * N_KV_HEADS + g) * HEAD_DIM);
                __builtin_prefetch(nk, 0, 3);
            }
            float s0 = 0, s1 = 0, s2 = 0, s3 = 0;
#pragma unroll 4
            for (int dd = 0; dd < HEAD_DIM; dd += 4) {
                float4 k4 = *(const float4*)(kptr + dd);
                float4 q0 = *(const float4*)(q_s + 0 * HEAD_DIM + dd);
                float4 q1 = *(const float4*)(q_s + 1 * HEAD_DIM + dd);
                float4 q2 = *(const float4*)(q_s + 2 * HEAD_DIM + dd);
                float4 q3 = *(const float4*)(q_s + 3 * HEAD_DIM + dd);
                s0 += q0.x * k4.x + q0.y * k4.y + q0.z * k4.z + q0.w * k4.w;
                s1 += q1.x * k4.x + q1.y * k4.y + q1.z * k4.z + q1.w * k4.w;
                s2 += q2.x * k4.x + q2.y * k4.y + q2.z * k4.z + q2.w * k4.w;
                s3 += q3.x * k4.x + q3.y * k4.y + q3.z * k4.z + q3.w * k4.w;
            }
            s0 *= scale; s1 *= scale; s2 *= scale; s3 *= scale;
            s_s[0 * TILE + pp] = s0; s_s[1 * TILE + pp] = s1;
            s_s[2 * TILE + pp] = s2; s_s[3 * TILE + pp] = s3;
            lm0 = fmaxf(lm0, s0); lm1 = fmaxf(lm1, s1);
            lm2 = fmaxf(lm2, s2); lm3 = fmaxf(lm3, s3);
        }
        red[0 * BLK + t] = lm0; red[1 * BLK + t] = lm1;
        red[2 * BLK + t] = lm2; red[3 * BLK + t] = lm3;
        __syncthreads();
        if (t < N_REP) {
            float m = -1e30f;
            for (int i = 0; i < BLK; ++i) m = fmaxf(m, red[t * BLK + i]);
            float nm = fmaxf(run_m[t], m);
            corr[t]   = __expf(run_m[t] - nm);
            tile_m[t] = nm;
            run_m[t]  = nm;
        }
        __syncthreads();

        // ---- exp + tile sums ----
        float ls0 = 0, ls1 = 0, ls2 = 0, ls3 = 0;
        const float m0 = tile_m[0], m1 = tile_m[1], m2 = tile_m[2], m3 = tile_m[3];
        for (int pp = t; pp < TILE; pp += BLK) {
            float e0 = __expf(s_s[0 * TILE + pp] - m0); s_s[0 * TILE + pp] = e0; ls0 += e0;
            float e1 = __expf(s_s[1 * TILE + pp] - m1); s_s[1 * TILE + pp] = e1; ls1 += e1;
            float e2 = __expf(s_s[2 * TILE + pp] - m2); s_s[2 * TILE + pp] = e2; ls2 += e2;
            float e3 = __expf(s_s[3 * TILE + pp] - m3); s_s[3 * TILE + pp] = e3; ls3 += e3;
        }
        red[0 * BLK + t] = ls0; red[1 * BLK + t] = ls1;
        red[2 * BLK + t] = ls2; red[3 * BLK + t] = ls3;
        __syncthreads();
        if (t < N_REP) {
            float s = 0.0f;
            for (int i = 0; i < BLK; ++i) s += red[t * BLK + i];
            run_l[t] = run_l[t] * corr[t] + s;
        }
        __syncthreads();

        // ---- accumulate prob * V ----
        acc0 *= corr[hp];
        acc1 *= corr[hp + 2];
        for (int pp = 0; pp < TILE; ++pp) {
            int pos = tile + pp;
            const float* vptr = (pos == KV_LEN - 1)
                ? vnew
                : (cache_v + (((size_t)b * KV_LEN + pos) * N_KV_HEADS + g) * HEAD_DIM);
            float v = vptr[d];
            acc0 += s_s[hp       * TILE + pp] * v;
            acc1 += s_s[(hp + 2) * TILE + pp] * v;
        }
        __syncthreads();
    }

    // write unnormalized partials
    const size_t base = ((size_t)bg * ASPLIT + sp) * N_REP;
    pacc[(base + hp)     * HEAD_DIM + d] = acc0;
    pacc[(base + hp + 2) * HEAD_DIM + d] = acc1;
    if (t < N_REP) {
        pm[base + t] = run_m[t];
        pl[base + t] = run_l[t];
    }
}

// ---------------------------------------------------------------------------
// Merge the ASPLIT partials per (b, g, h, d) with log-sum-exp rescaling.
// 16*8*4*128 = 65536 threads.
// ---------------------------------------------------------------------------
__global__ __launch_bounds__(256)
void attn_reduce(const float* __restrict__ pacc,
                 const float* __restrict__ pm,
                 const float* __restrict__ pl,
                 float* __restrict__ out)            // 16 x 4096
{
    int tid = blockIdx.x * blockDim.x + threadIdx.x; // < 65536
    int d   = tid & 127;
    int h   = (tid >> 7) & 3;
    int bg  = tid >> 9;
    int b   = bg >> 3;
    int g   = bg & 7;

    float m = -1e30f;
#pragma unroll
    for (int sp = 0; sp < ASPLIT; ++sp)
        m = fmaxf(m, pm[((size_t)bg * ASPLIT + sp) * N_REP + h]);

    float L = 0.0f, acc = 0.0f;
#pragma unroll
    for (int sp = 0; sp < ASPLIT; ++sp) {
        size_t base = ((size_t)bg * ASPLIT + sp) * N_REP + h;
        float w = __expf(pm[base] - m);
        L   += pl[base] * w;
        acc += pacc[base * HEAD_DIM + d] * w;
    }
    out[(size_t)b * DIM + (g * N_REP + h) * HEAD_DIM + d] = acc / L;
}

// ---------------------------------------------------------------------------
extern "C" void kernel_launch(void* const* d_in, const int* in_sizes, int n_in,
                              void* d_out, int out_size, void* d_ws, size_t ws_size,
                              hipStream_t stream) {
    const float* x       = (const float*)d_in[0];   // 16 x 1 x 4096
    const float* wq      = (const float*)d_in[1];   // 4096 x 4096
    const float* wk      = (const float*)d_in[2];   // 4096 x 1024
    const float* wv      = (const float*)d_in[3];   // 4096 x 1024
    const float* wo      = (const float*)d_in[4];   // 4096 x 4096
    const float* cache_k = (const float*)d_in[5];   // 16 x 4096 x 8 x 128
    const float* cache_v = (const float*)d_in[6];
    const float* fcos    = (const float*)d_in[7];   // 1 x 64
    const float* fsin    = (const float*)d_in[8];
    // d_in[9] = start_pos (4095, compile-time constant per reference)

    float* xq   = (float*)d_ws;                                    // 16*4096
    float* xk   = xq   + MAX_BS * DIM;                             // 16*1024
    float* xv   = xk   + MAX_BS * N_KV_HEADS * HEAD_DIM;           // 16*1024
    float* attn = xv   + MAX_BS * N_KV_HEADS * HEAD_DIM;           // 16*4096
    float* pacc = attn + MAX_BS * DIM;                             // 16*8*4*4*128
    float* pm   = pacc + (size_t)MAX_BS * N_KV_HEADS * ASPLIT * N_REP * HEAD_DIM;
    float* pl   = pm   + MAX_BS * N_KV_HEADS * ASPLIT * N_REP;
    float* out  = (float*)d_out;

    // QKV projections (WMMA f32 16x16x4, M = batch = 16, split-K x4 per block)
    gemm16_wmma<DIM><<<DIM / 16, 128, 0, stream>>>(x, wq, xq, DIM);
    gemm16_wmma<DIM><<<(N_KV_HEADS * HEAD_DIM) / 16, 128, 0, stream>>>(x, wk, xk, N_KV_HEADS * HEAD_DIM);
    gemm16_wmma<DIM><<<(N_KV_HEADS * HEAD_DIM) / 16, 128, 0, stream>>>(x, wv, xv, N_KV_HEADS * HEAD_DIM);

    // RoPE
    rope_q_kernel<<<(MAX_BS * N_HEADS * 64) / 256, 256, 0, stream>>>(xq, fcos, fsin);
    rope_k_kernel<<<(MAX_BS * N_KV_HEADS * 64) / 256, 256, 0, stream>>>(xk, fcos, fsin);

    // Flash-decode over 536 MB KV cache: 512 blocks (split-KV x4) + merge
    attn_partial<<<MAX_BS * N_KV_HEADS * ASPLIT, BLK, 0, stream>>>(
        xq, xk, xv, cache_k, cache_v, pacc, pm, pl);
    attn_reduce<<<(MAX_BS * N_KV_HEADS * N_REP * HEAD_DIM) / 256, 256, 0, stream>>>(
        pacc, pm, pl, attn);

    // Output projection
    gemm16_wmma<DIM><<<DIM / 16, 128, 0, stream>>>(attn, wo, out, DIM);
}